// QueryAndGroup_10574209482754
// MI455X (gfx1250) — compile-verified
//
#include <hip/hip_runtime.h>
#include <stdint.h>

// Problem constants (match reference)
#define BB      4
#define NN      8192
#define NPOINT  2048
#define CC      64
#define NSAMP   32
#define NCH     70                   // 3 + 3 + 64
#define NBLK    (NN / 32)            // 256 blocks of 32 points per centroid
#define NWORD   (NBLK / 32)          // 8 bitmap words per centroid
#define NSPLIT  4                    // phase-1 N-range splits
#define NLOC    (NN / NSPLIT)        // points per split (2048)
#define R2      (0.1f * 0.1f)        // jnp.float32(0.1)**2, exact f32 constant fold
#define TAU     (1e-3f)              // prune margin >> expansion rounding error
#define THR     (R2 + TAU)

typedef float v2f __attribute__((ext_vector_type(2)));
typedef float v8f __attribute__((ext_vector_type(8)));

// ---------------------------------------------------------------------------
// Cooperative async copy of `nbytes` (multiple of 256*16) into LDS.
// CDNA5 async-to-LDS path, tracked by ASYNCcnt.
// ---------------------------------------------------------------------------
__device__ __forceinline__ void async_copy_to_lds(float* sdst, const float* gsrc,
                                                  int nbytes, int tid) {
    unsigned lds_base = (unsigned)(uintptr_t)(__attribute__((address_space(3))) float*)sdst;
    for (int it = 0; it < nbytes / (256 * 16); ++it) {
        unsigned off = (unsigned)(it * 256 + tid) * 16u;
        unsigned lds_addr = lds_base + off;
        unsigned long long gaddr = (unsigned long long)(uintptr_t)((const char*)gsrc + off);
        asm volatile("global_load_async_to_lds_b128 %0, %1, off"
                     :: "v"(lds_addr), "v"(gaddr) : "memory");
    }
    asm volatile("s_wait_asynccnt 0" ::: "memory");
}

// ---------------------------------------------------------------------------
// Phase 1: WMMA pruning.  One wave per (16-centroid tile, N/4 point range).
// D[m][n] = |c_m|^2 - 2 c_m . q_n + |q_n|^2  via v_wmma_f32_16x16x4_f32.
// Emits, per centroid, 1 bit per 32-point block: "may contain a candidate".
// ---------------------------------------------------------------------------
__global__ void __launch_bounds__(256)
ballquery_prune(const float* __restrict__ xyz, const float* __restrict__ newxyz,
                unsigned* __restrict__ wsAny) {
    extern __shared__ float sxyz[];                  // NLOC*3 floats = 24 KB
    const int tid  = threadIdx.x;
    const int wave = tid >> 5;
    const int lane = tid & 31;
    const int b     = blockIdx.x >> 6;               // 64 blocks per batch
    const int split = (blockIdx.x >> 4) & (NSPLIT - 1);
    const int tileInB = ((blockIdx.x & 15) << 3) + wave;   // 0..127

    async_copy_to_lds(sxyz, xyz + ((size_t)b * NN + split * NLOC) * 3,
                      NLOC * 3 * 4, tid);
    __syncthreads();

    const int m    = lane & 15;                      // row (centroid) this lane feeds
    const int half = lane >> 4;                      // K placement half
    const int j0   = tileInB * 16;

    // A operand: row m = (-2cx, -2cy, -2cz, 1); K = {0,1} low half, {2,3} high half
    const float* cb = newxyz + ((size_t)b * NPOINT + j0 + m) * 3;
    float cx = cb[0], cy = cb[1], cz = cb[2];
    float nC = __fadd_rn(__fadd_rn(__fmul_rn(cx, cx), __fmul_rn(cy, cy)), __fmul_rn(cz, cz));
    v2f A;
    A.x = half ? (-2.0f * cz) : (-2.0f * cx);
    A.y = half ? 1.0f         : (-2.0f * cy);

    // C operand: C[r][*] = |c_row|^2, row = r + 8*half. Gather via shuffles.
    v8f Cm;
#pragma unroll
    for (int r = 0; r < 8; ++r)
        Cm[r] = __shfl(nC, r + 8 * half, 32);

    const int halfShift = (m >> 3) << 4;             // 0 or 16: which ballot half is row m
    unsigned acc = 0;                                // per-lane (row m) block bits

    for (int blkL = 0; blkL < NLOC / 32; ++blkL) {   // 64 local 32-point blocks
        unsigned balAcc[8] = {0, 0, 0, 0, 0, 0, 0, 0};
#pragma unroll
        for (int t = 0; t < 2; ++t) {
            const int n = (blkL * 2 + t) * 16 + m;   // local point index (column)
            const float* q = &sxyz[n * 3];
            float qx = q[0], qy = q[1], qz = q[2];
            float nQ = __fadd_rn(__fadd_rn(__fmul_rn(qx, qx), __fmul_rn(qy, qy)),
                                 __fmul_rn(qz, qz));
            v2f Bv;
            Bv.x = half ? qz : qx;                   // K=0 / K=2
            Bv.y = half ? nQ : qy;                   // K=1 / K=3

            v8f D = __builtin_amdgcn_wmma_f32_16x16x4_f32(
                false, A, false, Bv, (short)0, Cm, false, false);

#pragma unroll
            for (int r = 0; r < 8; ++r)
                balAcc[r] |= __builtin_amdgcn_ballot_w32(D[r] < THR);
        }
        // Per-lane select of this lane's row ballot (7 cndmasks, no SALU chain):
        unsigned s0 = (m & 1) ? balAcc[1] : balAcc[0];
        unsigned s1 = (m & 1) ? balAcc[3] : balAcc[2];
        unsigned s2 = (m & 1) ? balAcc[5] : balAcc[4];
        unsigned s3 = (m & 1) ? balAcc[7] : balAcc[6];
        unsigned t0 = (m & 2) ? s1 : s0;
        unsigned t1 = (m & 2) ? s3 : s2;
        unsigned v  = (m & 4) ? t1 : t0;
        unsigned rowAny = ((v >> halfShift) & 0xffffu) ? 1u : 0u;
        acc |= rowAny << (blkL & 31);
        if ((blkL & 31) == 31) {
            if (lane < 16)
                wsAny[((size_t)b * NPOINT + j0 + m) * NWORD
                      + split * (NLOC / 32 / 32) + (blkL >> 5)] = acc;
            acc = 0;
        }
    }
}

// ---------------------------------------------------------------------------
// Phase 2: exact ball query over flagged blocks only.  One wave handles 4
// centroids.  Scan-order compaction with ballot/popcount; identical semantics
// to the reference (first 32 in index order, pad with first hit, fps slot 0).
// ---------------------------------------------------------------------------
__global__ void __launch_bounds__(256)
ballquery_exact(const float* __restrict__ xyz, const float* __restrict__ newxyz,
                const int* __restrict__ fps, const unsigned* __restrict__ wsAny,
                int* __restrict__ wsIdx) {
    extern __shared__ float sxyz[];                  // NN*3 floats = 96 KB
    const int tid  = threadIdx.x;
    const int wave = tid >> 5;
    const int lane = tid & 31;
    const int b     = blockIdx.x >> 6;               // 64 blocks per batch
    const int group = blockIdx.x & 63;               // 32 centroids per block

    async_copy_to_lds(sxyz, xyz + (size_t)b * (NN * 3), NN * 3 * 4, tid);
    __syncthreads();

    for (int c = 0; c < 4; ++c) {
        const int j = group * 32 + wave * 4 + c;     // centroid in batch
        const float* cb = newxyz + ((size_t)b * NPOINT + j) * 3;
        const float cx = cb[0], cy = cb[1], cz = cb[2];
        int* out = wsIdx + ((size_t)b * NPOINT + j) * (NSAMP + 1);
        if (lane == 0) out[0] = fps[(size_t)b * NPOINT + j];

        const unsigned* anyw = wsAny + ((size_t)b * NPOINT + j) * NWORD;
        int cnt = 0, first = 0;
        bool haveFirst = false;

        for (int w = 0; w < NWORD && cnt < NSAMP; ++w) {
            unsigned aw = __builtin_amdgcn_readfirstlane(anyw[w]);
            while (aw && cnt < NSAMP) {
                const int bb = __builtin_ctz(aw);
                aw &= aw - 1u;
                const int blk = w * 32 + bb;
                const int n = blk * 32 + lane;
                const float* q = &sxyz[n * 3];
                // exact, contraction-free d2 matching jnp: (dx^2 + dy^2) + dz^2
                float dx = __fsub_rn(q[0], cx);
                float dy = __fsub_rn(q[1], cy);
                float dz = __fsub_rn(q[2], cz);
                float d2 = __fadd_rn(__fadd_rn(__fmul_rn(dx, dx), __fmul_rn(dy, dy)),
                                     __fmul_rn(dz, dz));
                const bool pred = d2 < R2;
                const unsigned msk = __builtin_amdgcn_ballot_w32(pred);
                if (msk) {
                    if (!haveFirst) { first = blk * 32 + __builtin_ctz(msk); haveFirst = true; }
                    const int rank = __builtin_popcount(msk & ((1u << lane) - 1u));
                    const int slot = cnt + rank;
                    if (pred && slot < NSAMP) out[1 + slot] = n;
                    cnt += __builtin_popcount(msk);
                }
            }
        }
        if (cnt > NSAMP) cnt = NSAMP;
        const int pad = haveFirst ? first : 0;
        for (int s = cnt + lane; s < NSAMP; s += 32) out[1 + s] = pad;
    }
}

// ---------------------------------------------------------------------------
// Phase 3: gather + center + concat.  One thread per (b, j, s) sample slot;
// each thread emits all 70 channels.  The idx load is fully coalesced (its
// layout equals the thread index), index math is amortized 70x, stores stay
// coalesced (lanes cover consecutive s), feature gathers are L2-resident.
// ---------------------------------------------------------------------------
__global__ void __launch_bounds__(256)
group_kernel(const float* __restrict__ xyz, const float* __restrict__ newxyz,
             const float* __restrict__ feat, const int* __restrict__ wsIdx,
             float* __restrict__ out, int total) {
    const int g = blockIdx.x * 256 + threadIdx.x;    // (b*NPOINT + j)*33 + s
    if (g >= total) return;
    const int s = g % (NSAMP + 1);
    const int bj = g / (NSAMP + 1);                  // b*NPOINT + j
    const int j = bj & (NPOINT - 1);
    const int b = bj >> 11;                          // NPOINT = 2048 = 1<<11

    const int idx = wsIdx[g];                        // coalesced

    // Centered xyz (channels 0-2, duplicated as 3-5)
    const float* p = xyz + ((size_t)b * NN + idx) * 3;
    const float* q = newxyz + (size_t)bj * 3;
    const float dx = __fsub_rn(p[0], q[0]);
    const float dy = __fsub_rn(p[1], q[1]);
    const float dz = __fsub_rn(p[2], q[2]);

    const size_t str = (size_t)NPOINT * (NSAMP + 1);             // channel stride
    float* o = out + ((size_t)b * NCH * NPOINT) * (NSAMP + 1)    // batch base
                   + (size_t)j * (NSAMP + 1) + s;
    o[0]       = dx;  o[str]     = dy;  o[2 * str] = dz;
    o[3 * str] = dx;  o[4 * str] = dy;  o[5 * str] = dz;

    // Feature channels 6..69: one gather + one coalesced store per channel
    const float* fp = feat + (size_t)b * CC * NN + idx;
    float* o2 = o + 6 * str;
#pragma unroll 8
    for (int c = 0; c < CC; ++c)
        o2[(size_t)c * str] = fp[(size_t)c * NN];
}

// ---------------------------------------------------------------------------
extern "C" void kernel_launch(void* const* d_in, const int* in_sizes, int n_in,
                              void* d_out, int out_size, void* d_ws, size_t ws_size,
                              hipStream_t stream) {
    const float* xyz    = (const float*)d_in[0];   // (B, N, 3)
    const float* newxyz = (const float*)d_in[1];   // (B, NPOINT, 3)
    const float* feat   = (const float*)d_in[2];   // (B, C, N)
    const int*   fps    = (const int*)d_in[3];     // (B, NPOINT)
    float* out = (float*)d_out;                    // (B, 70, NPOINT, 33)

    // Workspace: block-any bitmap (256 KB) then idx table (1.03 MB)
    unsigned* wsAny = (unsigned*)d_ws;
    int* wsIdx = (int*)((char*)d_ws + (size_t)BB * NPOINT * NWORD * sizeof(unsigned));

    // Phase 1: WMMA prune — 2048 waves, one per (16-centroid tile, N/4 range)
    ballquery_prune<<<BB * 16 * NSPLIT, 256, (size_t)NLOC * 3 * 4, stream>>>(xyz, newxyz, wsAny);

    // Phase 2: exact scan-order ball query — one wave per 4 centroids
    ballquery_exact<<<BB * (NPOINT / 32), 256, (size_t)NN * 3 * 4, stream>>>(
        xyz, newxyz, fps, wsAny, wsIdx);

    // Phase 3: gather/center/concat — one thread per (b, j, s), 70 outputs each
    const int total3 = BB * NPOINT * (NSAMP + 1);
    group_kernel<<<(total3 + 255) / 256, 256, 0, stream>>>(xyz, newxyz, feat, wsIdx, out, total3);
}